// Program_72902774882571
// MI455X (gfx1250) — compile-verified
//
#include <hip/hip_runtime.h>

// -------------------------------------------------------------------------
// MI455X (gfx1250) streaming kernel.
// Roofline: 16B in + 4B out per row, ~100MB total -> ~4.3us at 23.3TB/s HBM.
// Data path: CDNA5 async global->LDS (ASYNCcnt) 4-deep pipeline per wave.
// Compute: per-lane weight select (cndmask) + 12 FMA + relu + fast sigmoid.
// VALU trimmed via two uniform pre-folds:
//   (1) sel-term of layer 1 folded into per-net bias (sel is exactly 0/1/2
//       when that net is selected),
//   (2) -log2(e) folded into layer-2 weights so sigmoid = rcp(1+exp2(z')).
// No WMMA: K=3 per-row matmul, memory-bound — WMMA cannot move the floor.
// -------------------------------------------------------------------------

#define BLOCK  256
#define STAGES 4
#define ITERS  8

__device__ __forceinline__ float fast_exp2(float a) {
#if __has_builtin(__builtin_amdgcn_exp2f)
  return __builtin_amdgcn_exp2f(a);
#else
  return exp2f(a);
#endif
}

__device__ __forceinline__ float fast_rcp(float a) {
#if __has_builtin(__builtin_amdgcn_rcpf)
  return __builtin_amdgcn_rcpf(a);
#else
  return 1.0f / a;
#endif
}

// One 16-byte async global->LDS copy (GLOBAL_LOAD_ASYNC_TO_LDS_B128, GVS mode:
// saddr base + 32-bit vgpr byte offset). Tracked by ASYNCcnt.
__device__ __forceinline__ void async_load_row(const void* base, unsigned byte_off,
                                               unsigned lds_byte_addr) {
  asm volatile("global_load_async_to_lds_b128 %0, %1, %2"
               :
               : "v"(lds_byte_addr), "v"(byte_off), "s"(base)
               : "memory");
}

__device__ __forceinline__ void wait_asynccnt_le(int n) {
  // n is compile-time constant after full unroll; dead branches fold away.
  if (n <= 0)      asm volatile("s_wait_asynccnt 0" ::: "memory");
  else if (n == 1) asm volatile("s_wait_asynccnt 1" ::: "memory");
  else if (n == 2) asm volatile("s_wait_asynccnt 2" ::: "memory");
  else             asm volatile("s_wait_asynccnt 3" ::: "memory");
}

__device__ __forceinline__ void wait_dscnt0() {
  asm volatile("s_wait_dscnt 0" ::: "memory");
}

__global__ __launch_bounds__(BLOCK) void mlp3_select_kernel(
    const float4* __restrict__ x,
    const float* __restrict__ Ws1, const float* __restrict__ bs1,
    const float* __restrict__ Ws2, const float* __restrict__ bs2,
    const float* __restrict__ Wu1, const float* __restrict__ bu1,
    const float* __restrict__ Wu2, const float* __restrict__ bu2,
    const float* __restrict__ Wd1, const float* __restrict__ bd1,
    const float* __restrict__ Wd2, const float* __restrict__ bd2,
    float* __restrict__ out, unsigned n) {
  __shared__ float4 stage[STAGES * BLOCK];

  const unsigned tid = threadIdx.x;
  const unsigned T   = gridDim.x * BLOCK;           // total threads
  const unsigned g   = blockIdx.x * BLOCK + tid;    // global thread id

  const float NL2E = -1.44269504088896340736f;      // -log2(e)

  // ---- Uniform preamble: load + fold weights (amortized over 256 rows/wave).
  // a0 = W1 row 0, a2 = W1 row 2, bb = b1 + sel_val*W1 row 1 (sel fold),
  // w2 = -log2e * W2, c2 = -log2e * b2 (sigmoid scale fold).
  float a0s[4], a2s[4], bbs[4], w2s[4];
  float a0u[4], a2u[4], bbu[4], w2u[4];
  float a0d[4], a2d[4], bbd[4], w2d[4];
#pragma unroll
  for (int j = 0; j < 4; ++j) {
    a0s[j] = Ws1[j];      a2s[j] = Ws1[8 + j];
    a0u[j] = Wu1[j];      a2u[j] = Wu1[8 + j];
    a0d[j] = Wd1[j];      a2d[j] = Wd1[8 + j];
    bbs[j] = bs1[j];                                    // sel = 0
    bbu[j] = bu1[j] + Wu1[4 + j];                       // sel = 1
    bbd[j] = fmaf(2.0f, Wd1[4 + j], bd1[j]);            // sel = 2
    w2s[j] = NL2E * Ws2[j];
    w2u[j] = NL2E * Wu2[j];
    w2d[j] = NL2E * Wd2[j];
  }
  const float c2s = NL2E * bs2[0];
  const float c2u = NL2E * bu2[0];
  const float c2d = NL2E * bd2[0];

  const unsigned lds_base = (unsigned)(size_t)(&stage[0]);

  // ---- Prologue: fill the pipeline (uniform issuance keeps ASYNCcnt exact).
#pragma unroll
  for (int k = 0; k < STAGES; ++k) {
    unsigned row = g + (unsigned)k * T;
    row = row < n ? row : (n - 1u);                 // clamp: always issue
    async_load_row(x, row * 16u, lds_base + ((unsigned)k * BLOCK + tid) * 16u);
  }

  // ---- Main pipelined loop.
#pragma unroll
  for (int k = 0; k < ITERS; ++k) {
    const int pend = (ITERS - 1 - k) < (STAGES - 1) ? (ITERS - 1 - k) : (STAGES - 1);
    wait_asynccnt_le(pend);                         // oldest async copy done

    const int slot = k % STAGES;
    float4 v = stage[slot * BLOCK + tid];           // ds_load_b128
    wait_dscnt0();                                  // v in regs before slot reuse

    if (k + STAGES < ITERS) {                       // refill the slot
      unsigned row2 = g + (unsigned)(k + STAGES) * T;
      row2 = row2 < n ? row2 : (n - 1u);
      async_load_row(x, row2 * 16u, lds_base + ((unsigned)slot * BLOCK + tid) * 16u);
    }

    const unsigned row = g + (unsigned)k * T;
    if (row < n) {
      const float x0 = v.x, sl = v.y, x2 = v.z;     // x3 = (x0, sel, x2)
      const bool m0 = (sl == 0.0f);
      const bool m1 = (sl == 1.0f);
      const bool m2 = (sl == 2.0f);
      auto pick = [&](float a, float b, float c) {
        float r = m1 ? b : a;
        return m2 ? c : r;
      };

      // Layer 1 (sel-term pre-folded): h_j = relu(B[j] + x0*A0[j] + x2*A2[j])
      float h[4];
#pragma unroll
      for (int j = 0; j < 4; ++j) {
        float hj = pick(bbs[j], bbu[j], bbd[j]);
        hj = fmaf(x0, pick(a0s[j], a0u[j], a0d[j]), hj);
        hj = fmaf(x2, pick(a2s[j], a2u[j], a2d[j]), hj);
        h[j] = fmaxf(hj, 0.0f);
      }

      // Layer 2 with -log2e pre-scaled weights: z' = -log2e * (h.W2 + b2)
      float z = pick(c2s, c2u, c2d);
#pragma unroll
      for (int j = 0; j < 4; ++j)
        z = fmaf(h[j], pick(w2s[j], w2u[j], w2d[j]), z);

      const float sig = fast_rcp(1.0f + fast_exp2(z));  // 1/(1+e^-zorig)
      out[row] = (m0 | m1 | m2) ? sig : 0.0f;           // jnp.where chain
    }
  }
}

extern "C" void kernel_launch(void* const* d_in, const int* in_sizes, int n_in,
                              void* d_out, int out_size, void* d_ws, size_t ws_size,
                              hipStream_t stream) {
  (void)n_in; (void)d_ws; (void)ws_size; (void)out_size;
  const unsigned B = (unsigned)(in_sizes[0] / 4);   // x is (B, 4) float32

  const float4* x   = (const float4*)d_in[0];
  const float* Ws1  = (const float*)d_in[1];
  const float* bs1_ = (const float*)d_in[2];
  const float* Ws2  = (const float*)d_in[3];
  const float* bs2_ = (const float*)d_in[4];
  const float* Wu1  = (const float*)d_in[5];
  const float* bu1_ = (const float*)d_in[6];
  const float* Wu2  = (const float*)d_in[7];
  const float* bu2_ = (const float*)d_in[8];
  const float* Wd1  = (const float*)d_in[9];
  const float* bd1_ = (const float*)d_in[10];
  const float* Wd2  = (const float*)d_in[11];
  const float* bd2_ = (const float*)d_in[12];
  float* out = (float*)d_out;

  const unsigned rows_per_blk = BLOCK * ITERS;
  const unsigned nblk = (B + rows_per_blk - 1) / rows_per_blk;  // 2048 for B=4M

  mlp3_select_kernel<<<nblk, BLOCK, 0, stream>>>(
      x, Ws1, bs1_, Ws2, bs2_, Wu1, bu1_, Wu2, bu2_, Wd1, bd1_, Wd2, bd2_,
      out, B);
}